// TPNet_3882650437025
// MI455X (gfx1250) — compile-verified
//
#include <hip/hip_runtime.h>
#include <hip/hip_bf16.h>
#include <math.h>

// ---- problem constants (match reference) ----
#define MM     3          // decay scales
#define KK1    3          // layers + 1
#define NN     100000     // nodes
#define DDIM   128        // projection dim
#define EE     500000     // edges
#define BB     200000     // query pairs
#define PFD    36         // pairwise feature dim (2*K1)^2
#define HID    144        // 4*PF

typedef __attribute__((ext_vector_type(2))) float v2f;
typedef __attribute__((ext_vector_type(4))) float v4f;
typedef __attribute__((ext_vector_type(8))) float v8f;

// fused state elements and ws layout (in floats)
#define FUSED_ELEMS ((long)NN * KK1 * DDIM)       // 38,400,000
#define COEF_OFF    FUSED_ELEMS
#define FEAT_OFF    (FUSED_ELEMS + 32)

__device__ __forceinline__ v8f wmma4(v2f a, v2f b, v8f c) {
    // V_WMMA_F32_16X16X4_F32: D = A(16x4) x B(4x16) + C(16x16), all f32
    return __builtin_amdgcn_wmma_f32_16x16x4_f32(
        false, a, false, b, (short)0, c, false, false);
}

// ---------------------------------------------------------------------------
// Kernel 0: compute scalar coefficient table on-device.
//   coef[m*3+k]      = W[k,m] * base_m^k          (init scaling)
//   coef[9+m*3+k]    = W[k,m] * base_m^(k-1)      (edge-message scaling, k>=1)
//   coef[18]         = next_time (= times[E-1])
// W = softmax(lambda_weights, axis=1); base_m = exp(-lam_m * next_time)
// ---------------------------------------------------------------------------
__global__ void k_setup(const float* __restrict__ lamw,
                        const int* __restrict__ times,
                        float* __restrict__ coef) {
    if (threadIdx.x != 0 || blockIdx.x != 0) return;
    const float lams[3] = {1e-6f, 1e-4f, 1e-2f};
    float T = (float)times[EE - 1];
    float W[3][3];
    for (int k = 0; k < 3; ++k) {
        float mx = lamw[k * 3 + 0];
        for (int m = 1; m < 3; ++m) mx = fmaxf(mx, lamw[k * 3 + m]);
        float e[3], s = 0.f;
        for (int m = 0; m < 3; ++m) { e[m] = __expf(lamw[k * 3 + m] - mx); s += e[m]; }
        for (int m = 0; m < 3; ++m) W[k][m] = e[m] / s;
    }
    for (int m = 0; m < 3; ++m) {
        float base = __expf(-lams[m] * T);
        float bp = 1.f;       // base^k
        float bpm = 1.f;      // base^(k-1), defined for k>=1
        for (int k = 0; k < 3; ++k) {
            coef[m * 3 + k] = W[k][m] * bp;
            coef[9 + m * 3 + k] = (k >= 1) ? W[k][m] * bpm : 0.f;
            if (k >= 1) bpm *= base;
            bp *= base;
        }
    }
    coef[18] = T;
}

// ---------------------------------------------------------------------------
// Kernel 1: fused init.  fused[n,k,d] = sum_m coefA[m][k] * proj[m,k,n,d]
// One thread per 4 dims (v4 loads/stores, fully coalesced).
// ---------------------------------------------------------------------------
__global__ __launch_bounds__(256)
void k_init(const float* __restrict__ proj,
            const float* __restrict__ coef,
            float* __restrict__ fused) {
    long tid = (long)blockIdx.x * blockDim.x + threadIdx.x;
    long base = tid * 4;
    if (base >= FUSED_ELEMS) return;
    int d = (int)(base % DDIM);
    long t = base / DDIM;
    int k = (int)(t % KK1);
    long n = t / KK1;
    const long plane = (long)KK1 * NN * DDIM;
    long off = (long)k * NN * DDIM + n * DDIM + d;
    v4f acc = {0.f, 0.f, 0.f, 0.f};
    for (int m = 0; m < 3; ++m) {
        v4f p = *(const v4f*)(proj + (long)m * plane + off);
        float c = coef[m * 3 + k];
        acc += c * p;
    }
    *(v4f*)(fused + base) = acc;
}

// ---------------------------------------------------------------------------
// Kernel 2: edge scatter.  One wave (32 lanes) per edge; 4 dims per lane.
// For k in {1,2}:
//   fused[src,k] += sum_m c[m,k]*tw_m(e) * proj[m,k-1][dst]
//   fused[dst,k] += sum_m c[m,k]*tw_m(e) * proj[m,k-1][src]
// Accumulate scales in registers, then 16 hardware f32 atomics per lane.
// ---------------------------------------------------------------------------
__global__ __launch_bounds__(256)
void k_edges(const float* __restrict__ proj,
             const int* __restrict__ src_ids,
             const int* __restrict__ dst_ids,
             const int* __restrict__ times,
             const float* __restrict__ coef,
             float* __restrict__ fused) {
    int wave = (int)(((long)blockIdx.x * blockDim.x + threadIdx.x) >> 5);
    int lane = threadIdx.x & 31;
    if (wave >= EE) return;
    const int s = src_ids[wave];
    const int d = dst_ids[wave];
    const float t = (float)times[wave];
    const float T = coef[18];
    const float lams[3] = {1e-6f, 1e-4f, 1e-2f};
    const long plane = (long)KK1 * NN * DDIM;
    const long ldplane = (long)NN * DDIM;
    const int d4 = lane * 4;

    v4f aS1 = {0,0,0,0}, aS2 = {0,0,0,0}, aD1 = {0,0,0,0}, aD2 = {0,0,0,0};
    for (int m = 0; m < 3; ++m) {
        float tw = __expf(-lams[m] * (T - t));
        float c1 = coef[9 + m * 3 + 1] * tw;
        float c2 = coef[9 + m * 3 + 2] * tw;
        const float* pm = proj + (long)m * plane;
        v4f p0d = *(const v4f*)(pm + (long)d * DDIM + d4);
        v4f p0s = *(const v4f*)(pm + (long)s * DDIM + d4);
        v4f p1d = *(const v4f*)(pm + ldplane + (long)d * DDIM + d4);
        v4f p1s = *(const v4f*)(pm + ldplane + (long)s * DDIM + d4);
        aS1 += c1 * p0d;  aS2 += c2 * p1d;
        aD1 += c1 * p0s;  aD2 += c2 * p1s;
    }
    float* fS1 = fused + (long)s * (KK1 * DDIM) + 1 * DDIM + d4;
    float* fS2 = fused + (long)s * (KK1 * DDIM) + 2 * DDIM + d4;
    float* fD1 = fused + (long)d * (KK1 * DDIM) + 1 * DDIM + d4;
    float* fD2 = fused + (long)d * (KK1 * DDIM) + 2 * DDIM + d4;
    #pragma unroll
    for (int j = 0; j < 4; ++j) {
        unsafeAtomicAdd(fS1 + j, aS1[j]);
        unsafeAtomicAdd(fS2 + j, aS2[j]);
        unsafeAtomicAdd(fD1 + j, aD1[j]);
        unsafeAtomicAdd(fD2 + j, aD2[j]);
    }
}

// ---------------------------------------------------------------------------
// Kernel 3: pairwise Gram features.  One wave per query pair.
// rp = [fused[qs] (3x128); fused[qd] (3x128)]; feat = log1p(relu(rp rp^T)).
// 21 unique dot products; 5-step shfl_xor butterfly over the wave32.
// ---------------------------------------------------------------------------
__global__ __launch_bounds__(256)
void k_pairs(const float* __restrict__ fused,
             const int* __restrict__ q_src,
             const int* __restrict__ q_dst,
             float* __restrict__ feat) {
    int wave = (int)(((long)blockIdx.x * blockDim.x + threadIdx.x) >> 5);
    int lane = threadIdx.x & 31;
    if (wave >= BB) return;
    const int a = q_src[wave];
    const int b = q_dst[wave];
    const int d4 = lane * 4;
    v4f r[6];
    #pragma unroll
    for (int k = 0; k < 3; ++k) {
        r[k]     = *(const v4f*)(fused + (long)a * (KK1 * DDIM) + k * DDIM + d4);
        r[3 + k] = *(const v4f*)(fused + (long)b * (KK1 * DDIM) + k * DDIM + d4);
    }
    float p[21];
    int idx = 0;
    #pragma unroll
    for (int i = 0; i < 6; ++i)
        #pragma unroll
        for (int j = i; j < 6; ++j) {
            v4f pr = r[i] * r[j];
            p[idx++] = pr.x + pr.y + pr.z + pr.w;
        }
    #pragma unroll
    for (int off = 16; off >= 1; off >>= 1)
        #pragma unroll
        for (int q = 0; q < 21; ++q)
            p[q] += __shfl_xor(p[q], off, 32);
    // all lanes now hold full sums; write 36 entries (symmetric expand)
    for (int e = lane; e < PFD; e += 32) {
        int i = e / 6, j = e % 6;
        int ii = min(i, j), jj = max(i, j);
        int pidx = ii * 6 - (ii * (ii - 1)) / 2 + (jj - ii);
        float v = fmaxf(p[pidx], 0.f);
        feat[(long)wave * PFD + e] = log1pf(v);
    }
}

// ---------------------------------------------------------------------------
// Kernel 4: MLP via V_WMMA_F32_16X16X4_F32.  One wave = 16 query rows.
// All weights/biases staged in LDS once per block (41 KB); B-fragments are
// ds_load-fed so the wmma chain is not gated on global load latency.
// GEMM1: X[16x36] * w1[36x144] + b1 -> relu -> H (LDS, per-wave slab)
// GEMM2: H[16x144] * w2[144x36] + b2 -> out
// A layout (f32 16x4): lanes 0-15 hold K={k0,k0+1}, lanes 16-31 K={k0+2,k0+3}.
// B layout (f32 4x16): same K split, N = lane%16.  C: row=j+8*(lane/16), col=lane%16.
// Bias is folded into the accumulator init (bias is column-broadcast in C).
// ---------------------------------------------------------------------------
__global__ __launch_bounds__(128)
void k_mlp(const float* __restrict__ feat,
           const float* __restrict__ w1, const float* __restrict__ b1,
           const float* __restrict__ w2, const float* __restrict__ b2,
           float* __restrict__ out) {
    __shared__ float w1s[PFD * HID];      // 20.25 KB
    __shared__ float b1s[HID];
    __shared__ float w2s[HID * PFD];      // 20.25 KB
    __shared__ float b2s[PFD];
    __shared__ float Hs[4][16 * HID];     // 36 KB: per-wave 16x144 hidden slab

    // cooperative weight staging (block = 128 threads)
    for (int i = threadIdx.x; i < PFD * HID; i += 128) w1s[i] = w1[i];
    for (int i = threadIdx.x; i < HID * PFD; i += 128) w2s[i] = w2[i];
    if (threadIdx.x < HID) b1s[threadIdx.x] = b1[threadIdx.x];
    if (threadIdx.x < PFD) b2s[threadIdx.x] = b2[threadIdx.x];
    __syncthreads();

    const int wid  = threadIdx.x >> 5;
    const int lane = threadIdx.x & 31;
    const int tile = blockIdx.x * 4 + wid;
    if (tile >= BB / 16) return;          // after barrier: never divergent
    const int b0 = tile * 16;
    const int hi = lane >> 4;             // which K-half this lane holds
    const int lo = lane & 15;             // M (for A) / N (for B,C)

    // ---- GEMM1: preload all 9 A-fragments (reused across 9 N-tiles) ----
    v2f afrag[9];
    #pragma unroll
    for (int kk = 0; kk < 9; ++kk) {
        int k0 = kk * 4 + 2 * hi;
        afrag[kk] = *(const v2f*)(feat + (long)(b0 + lo) * PFD + k0);
    }
    float* Hw = Hs[wid];
    #pragma unroll
    for (int nt = 0; nt < 9; ++nt) {
        const int n0 = nt * 16;
        const float bias = b1s[n0 + lo];
        v8f c = {bias, bias, bias, bias, bias, bias, bias, bias};
        #pragma unroll
        for (int kk = 0; kk < 9; ++kk) {
            int k0 = kk * 4 + 2 * hi;
            v2f bf;
            bf.x = w1s[(k0 + 0) * HID + n0 + lo];
            bf.y = w1s[(k0 + 1) * HID + n0 + lo];
            c = wmma4(afrag[kk], bf, c);
        }
        #pragma unroll
        for (int j = 0; j < 8; ++j)
            Hw[(j + 8 * hi) * HID + n0 + lo] = fmaxf(c[j], 0.f);
    }
    // same-wave LDS RAW on the private H slab: compiler inserts s_wait_dscnt.

    // ---- GEMM2: 3 N-tiles (last has 4 valid cols), 36 K-steps ----
    #pragma unroll
    for (int nt = 0; nt < 3; ++nt) {
        const int n0 = nt * 16;
        const int col = n0 + lo;
        const int ccol = (col < PFD) ? col : 0;   // clamp lanes in the dead tail
        const float bias = b2s[ccol];
        v8f c = {bias, bias, bias, bias, bias, bias, bias, bias};
        #pragma unroll 6
        for (int kk = 0; kk < 36; ++kk) {
            int k0 = kk * 4 + 2 * hi;
            v2f af = *(const v2f*)(Hw + lo * HID + k0);
            v2f bf;
            bf.x = w2s[(k0 + 0) * PFD + ccol];
            bf.y = w2s[(k0 + 1) * PFD + ccol];
            c = wmma4(af, bf, c);
        }
        if (col < PFD) {
            #pragma unroll
            for (int j = 0; j < 8; ++j)
                out[(long)(b0 + j + 8 * hi) * PFD + col] = c[j];
        }
    }
}

// ---------------------------------------------------------------------------
extern "C" void kernel_launch(void* const* d_in, const int* in_sizes, int n_in,
                              void* d_out, int out_size, void* d_ws, size_t ws_size,
                              hipStream_t stream) {
    const float* proj = (const float*)d_in[0];
    const float* lamw = (const float*)d_in[1];
    const float* w1   = (const float*)d_in[2];
    const float* b1   = (const float*)d_in[3];
    const float* w2   = (const float*)d_in[4];
    const float* b2   = (const float*)d_in[5];
    const int* src    = (const int*)d_in[6];
    const int* dst    = (const int*)d_in[7];
    const int* times  = (const int*)d_in[8];
    const int* qs     = (const int*)d_in[9];
    const int* qd     = (const int*)d_in[10];
    float* out = (float*)d_out;

    float* ws    = (float*)d_ws;
    float* fused = ws;                  // [N, K1, DIM]  153.6 MB
    float* coef  = ws + COEF_OFF;       // 32 floats
    float* feat  = ws + FEAT_OFF;       // [B, 36]       28.8 MB

    // 0) scalar coefficient table
    k_setup<<<1, 32, 0, stream>>>(lamw, times, coef);
    // 1) fused init: 38.4M floats, 4 per thread
    {
        long threads = FUSED_ELEMS / 4;
        int blocks = (int)((threads + 255) / 256);
        k_init<<<blocks, 256, 0, stream>>>(proj, coef, fused);
    }
    // 2) edge scatter: one wave per edge
    {
        int blocks = (EE + 7) / 8;      // 8 waves per 256-thread block
        k_edges<<<blocks, 256, 0, stream>>>(proj, src, dst, times, coef, fused);
    }
    // 3) pairwise Gram features: one wave per pair
    {
        int blocks = (BB + 7) / 8;
        k_pairs<<<blocks, 256, 0, stream>>>(fused, qs, qd, feat);
    }
    // 4) WMMA MLP: one wave per 16 pairs, 4 waves per block
    {
        int tiles = BB / 16;            // 12500
        int blocks = (tiles + 3) / 4;
        k_mlp<<<blocks, 128, 0, stream>>>(feat, w1, b1, w2, b2, out);
    }
}